// Attention_EPGO_53266184405588
// MI455X (gfx1250) — compile-verified
//
#include <hip/hip_runtime.h>
#include <math.h>

typedef __attribute__((ext_vector_type(2))) float    v2f;
typedef __attribute__((ext_vector_type(8))) float    v8f;
typedef __attribute__((ext_vector_type(4))) unsigned v4u_t;
typedef __attribute__((ext_vector_type(8))) int      v8i_t;
typedef __attribute__((ext_vector_type(4))) int      v4i_t;

#define NB    4
#define DIM   256
#define NHEAD 8
#define NPIX  1024          // 32*32
#define QKV_OUT 512
#define HEAD_DIM 32
#define SCALE 0.25f         // KEY_DIM^-0.5

#ifndef __has_builtin
#define __has_builtin(x) 0
#endif
#if __has_builtin(__builtin_amdgcn_tensor_load_to_lds) && \
    __has_builtin(__builtin_amdgcn_s_wait_tensorcnt)
#define HAVE_TDM 1
#else
#define HAVE_TDM 0
#endif

static __device__ __forceinline__ unsigned key_of(float f) {
    unsigned u = __float_as_uint(f);
    return (u & 0x80000000u) ? ~u : (u | 0x80000000u);
}

// ---------------------------------------------------------------------------
// Gating MLP: g = sigmoid(g2 . relu(g1 . x + b1) + b2), one block per pixel
// ---------------------------------------------------------------------------
__global__ void k_gate(const float* __restrict__ x,
                       const float* __restrict__ g1_w, const float* __restrict__ g1_b,
                       const float* __restrict__ g2_w, const float* __restrict__ g2_b,
                       float* __restrict__ gate_ws) {
    __shared__ float red[128];
    int pix = blockIdx.x;                 // 0..4095
    int b = pix >> 10, n = pix & 1023;
    int t = threadIdx.x;                  // 0..127
    const float* xb = x + (size_t)b * DIM * NPIX + n;
    const float* wrow = g1_w + t * DIM;
    float h = g1_b[t];
    for (int c = 0; c < DIM; ++c) h += wrow[c] * xb[(size_t)c * NPIX];
    h = fmaxf(h, 0.0f);
    red[t] = g2_w[t] * h;
    __syncthreads();
    for (int off = 64; off > 0; off >>= 1) {
        if (t < off) red[t] += red[t + off];
        __syncthreads();
    }
    if (t == 0) gate_ws[pix] = 1.0f / (1.0f + __expf(-(red[0] + g2_b[0])));
}

// Deterministic fixed-order reduction -> dyn_k
__global__ void k_dynk(const float* __restrict__ gate_ws, int* __restrict__ dynk) {
    __shared__ float red[256];
    int t = threadIdx.x;
    float s = 0.0f;
    for (int i = t; i < NB * NPIX; i += 256) s += gate_ws[i];
    red[t] = s;
    __syncthreads();
    for (int off = 128; off > 0; off >>= 1) {
        if (t < off) red[t] += red[t + off];
        __syncthreads();
    }
    if (t == 0) {
        float mean = red[0] / (float)(NB * NPIX);
        int k = (int)floorf((float)NPIX * mean);
        if (k < 0) k = 0;
        if (k > NPIX) k = NPIX;
        dynk[0] = k;
    }
}

// ---------------------------------------------------------------------------
// Shared WMMA GEMM: out[b,o,n] = sum_c W[o,c]*in[b,c,n] + bias[o]
// Block = 4 waves = 64x16 output tile. The 256x16 B-panel is staged once in
// LDS (TDM async tensor load when available) and reused by all 4 waves.
// Row count = gridDim.x * 64 (512 for qkv, 256 for proj).
// ---------------------------------------------------------------------------
__global__ void k_gemm(const float* __restrict__ in, const float* __restrict__ w,
                       const float* __restrict__ bias, float* __restrict__ out) {
    __shared__ float Bt[DIM][16];         // 16 KB B-panel
    int wave = threadIdx.x >> 5, lane = threadIdx.x & 31;
    int i = lane & 15, half = lane >> 4;
    int o0 = blockIdx.x * 64 + wave * 16;
    int n0 = blockIdx.y * 16;
    int b  = blockIdx.z;
    int rows = gridDim.x * 64;
    const float* xb = in + (size_t)b * DIM * NPIX + n0;

#if HAVE_TDM
    if (wave == 0) {
        // Tensor DMA descriptor (2-D): tile 16 x 256 from a 1024 x 256 f32
        // tensor, row stride 1024 elements, into Bt (rows contiguous, 16 wide).
        unsigned lds_off = (unsigned)(size_t)(void*)&Bt[0][0];
        unsigned long long ga = (unsigned long long)(size_t)(const void*)xb;
        v4u_t g0 = { 1u,                                   // count=1 valid D#
                     lds_off,                              // lds_addr
                     (unsigned)ga,                         // global_addr[31:0]
                     (unsigned)((ga >> 32) & 0x01FFFFFFull) | (2u << 30) }; // type=2
        v8i_t g1 = { 0x00020000,      // data_size=4B, no multicast/pad/iterate
                     0x04000000,      // tensor_dim0 = 1024 (bits 63:48)
                     0x01000000,      // tensor_dim1 = 256  (bits 95:80)
                     0x00100000,      // tile_dim0  = 16    (bits 127:112)
                     256,             // tile_dim1  = 256
                     1024,            // tensor_dim0_stride = 1024
                     0, 0 };
        v4i_t gz = { 0, 0, 0, 0 };
#if defined(__clang_major__) && __clang_major__ >= 23
        v8i_t g4 = { 0, 0, 0, 0, 0, 0, 0, 0 };
        __builtin_amdgcn_tensor_load_to_lds(g0, g1, gz, gz, g4, 0);
#else
        __builtin_amdgcn_tensor_load_to_lds(g0, g1, gz, gz, 0);
#endif
        __builtin_amdgcn_s_wait_tensorcnt(0);
    }
#else
    for (int e = threadIdx.x; e < DIM * 16; e += 128)
        Bt[e >> 4][e & 15] = xb[(size_t)(e >> 4) * NPIX + (e & 15)];
#endif

    // Pull the whole 1 KB A-row toward the caches before the MMA loop
    // (unconditional -> no control flow inside the hot loop).
    const float* wrow = w + (size_t)(o0 + i) * DIM;
    __builtin_prefetch(wrow, 0, 0);
    __builtin_prefetch(wrow + 64, 0, 0);
    __builtin_prefetch(wrow + 128, 0, 0);
    __builtin_prefetch(wrow + 192, 0, 0);

    __syncthreads();

    v8f acc = {};
    for (int k = 0; k < DIM; k += 4) {
        int ka = k + half * 2;
        v2f a, bm;
        a.x  = wrow[ka];
        a.y  = wrow[ka + 1];
        bm.x = Bt[ka][i];
        bm.y = Bt[ka + 1][i];
        acc = __builtin_amdgcn_wmma_f32_16x16x4_f32(false, a, false, bm,
                                                    (short)0, acc, false, false);
    }
    float* outp = out + (size_t)b * rows * NPIX;
    int mr0 = o0 + half * 8;
    for (int r = 0; r < 8; ++r)
        outp[(size_t)(mr0 + r) * NPIX + n0 + i] = acc[r] + bias[mr0 + r];
}

// ---------------------------------------------------------------------------
// Fused attention: scores (WMMA) -> exact top-k (stable index ties, register
// resident) -> softmax -> P @ V^T (WMMA). One block (4 waves) = 16 queries of
// one (b,h). Attention matrix lives only in 64KB LDS.
// ---------------------------------------------------------------------------
__global__ void k_attn(const float* __restrict__ qkv, const int* __restrict__ dynk_p,
                       float* __restrict__ out_ws) {
    __shared__ float S[16][1024];
    int wave = threadIdx.x >> 5;
    int lane = threadIdx.x & 31;
    int i = lane & 15, half = lane >> 4;
    int m0 = blockIdx.x * 16;
    int h = blockIdx.y, b = blockIdx.z;
    int dynk = dynk_p[0];

    const float* base = qkv + (size_t)b * QKV_OUT * NPIX + (size_t)h * 64 * NPIX;
    const float* qptr = base;                   // rows 0..15  (d x n)
    const float* kptr = base + 16 * NPIX;       // rows 16..31
    const float* vptr = base + 32 * NPIX;       // rows 32..63

    // ---- Phase 1: S[m,n] = SCALE * sum_d q[d,m0+m] k[d,n]  (WMMA, K=16)
    // The q A-fragments depend only on kd: load once, reuse for all 16 column
    // tiles handled by this wave.
    v2f afrag[4];
#pragma unroll
    for (int t = 0; t < 4; ++t) {
        int ka = t * 4 + half * 2;
        afrag[t].x = qptr[(size_t)ka * NPIX + m0 + i];
        afrag[t].y = qptr[(size_t)(ka + 1) * NPIX + m0 + i];
    }
    for (int j = wave; j < 64; j += 4) {
        v8f acc = {};
#pragma unroll
        for (int t = 0; t < 4; ++t) {
            int ka = t * 4 + half * 2;
            v2f bm;
            bm.x = kptr[(size_t)ka * NPIX + j * 16 + i];
            bm.y = kptr[(size_t)(ka + 1) * NPIX + j * 16 + i];
            acc = __builtin_amdgcn_wmma_f32_16x16x4_f32(false, afrag[t], false, bm,
                                                        (short)0, acc, false, false);
        }
        for (int r = 0; r < 8; ++r)
            S[half * 8 + r][j * 16 + i] = acc[r] * SCALE;
    }
    __syncthreads();

    // ---- Phase 2: per-row top-dynk mask + softmax. Wave w owns rows 4w..4w+3;
    // each lane caches its 32 row elements in registers (one LDS read + one
    // LDS write per element total); all reductions intra-wave (wave32).
    unsigned lanemask_lt = (1u << lane) - 1u;
    for (int rr = 0; rr < 4; ++rr) {
        float* s = &S[wave * 4 + rr][0];
        float vals[32];
        unsigned keys[32];
#pragma unroll
        for (int c = 0; c < 32; ++c) {
            vals[c] = s[c * 32 + lane];
            keys[c] = key_of(vals[c]);
        }
        // dynk-th largest key via 32-step bitwise binary search (registers)
        unsigned T = 0u;
        for (int bit = 31; bit >= 0; --bit) {
            unsigned cand = T | (1u << bit);
            int cnt = 0;
#pragma unroll
            for (int c = 0; c < 32; ++c) cnt += (keys[c] >= cand) ? 1 : 0;
            for (int off = 16; off > 0; off >>= 1)
                cnt += __shfl_xor(cnt, off, 32);
            if (cnt >= dynk) T = cand;
        }
        int cgt = 0;
        float mval = -3.402823466e+38f;
#pragma unroll
        for (int c = 0; c < 32; ++c) {
            cgt += (keys[c] > T) ? 1 : 0;
            mval = fmaxf(mval, vals[c]);
        }
        for (int off = 16; off > 0; off >>= 1) {
            cgt += __shfl_xor(cgt, off, 32);
            mval = fmaxf(mval, __shfl_xor(mval, off, 32));
        }
        int rem = dynk - cgt;   // how many ==T survive (earliest indices win)
        float lsum = 0.0f;
        int eqseen = 0;
#pragma unroll
        for (int c = 0; c < 32; ++c) {
            bool eq = (keys[c] == T);
            unsigned bal32 = (unsigned)__ballot(eq);
            int myrank = eqseen + __popc(bal32 & lanemask_lt);
            bool keep = (keys[c] > T) || (eq && (myrank < rem));
            float p = keep ? __expf(vals[c] - mval) : 0.0f;
            lsum += p;
            vals[c] = p;
            eqseen += __popc(bal32);
        }
        for (int off = 16; off > 0; off >>= 1) lsum += __shfl_xor(lsum, off, 32);
        float inv = 1.0f / lsum;
#pragma unroll
        for (int c = 0; c < 32; ++c) s[c * 32 + lane] = vals[c] * inv;
    }
    __syncthreads();

    // ---- Phase 3: out[m,d] = sum_n P[m,n] v[d,n]  (WMMA, K=1024). Waves 0/1
    // take the two 16-wide d tiles (wave-uniform branch keeps EXEC all-1).
    if (wave < 2) {
        int d0 = wave * 16;
        v8f acc = {};
        for (int kd = 0; kd < 1024; kd += 4) {
            int ka = kd + half * 2;
            v2f a, bm;
            a.x  = S[i][ka];
            a.y  = S[i][ka + 1];
            bm.x = vptr[(size_t)(d0 + i) * NPIX + ka];
            bm.y = vptr[(size_t)(d0 + i) * NPIX + ka + 1];
            acc = __builtin_amdgcn_wmma_f32_16x16x4_f32(false, a, false, bm,
                                                        (short)0, acc, false, false);
        }
        // D[m=r+8*half, n=i] -> out_ws[b][h*32 + d0 + i][m0 + m]
        float* op = out_ws + (size_t)b * DIM * NPIX
                           + (size_t)(h * HEAD_DIM + d0 + i) * NPIX
                           + m0 + half * 8;
        for (int r = 0; r < 8; ++r) op[r] = acc[r];
    }
}

// ---------------------------------------------------------------------------
// Depthwise 3x3 PE conv on V image, accumulated into attention output
// ---------------------------------------------------------------------------
__global__ void k_pe(const float* __restrict__ qkv, const float* __restrict__ pe_w,
                     const float* __restrict__ pe_b, float* __restrict__ out_ws) {
    int idx = blockIdx.x * blockDim.x + threadIdx.x;
    if (idx >= NB * DIM * NPIX) return;
    int xx = idx & 31, yy = (idx >> 5) & 31, c = (idx >> 10) & 255, b = idx >> 18;
    const float* vimg = qkv + (size_t)b * QKV_OUT * NPIX
                            + (size_t)((c >> 5) * 64 + 32 + (c & 31)) * NPIX;
    const float* wk = pe_w + c * 9;
    float acc = pe_b[c];
    for (int dy = -1; dy <= 1; ++dy)
        for (int dx = -1; dx <= 1; ++dx) {
            int y2 = yy + dy, x2 = xx + dx;
            if (y2 >= 0 && y2 < 32 && x2 >= 0 && x2 < 32)
                acc += vimg[y2 * 32 + x2] * wk[(dy + 1) * 3 + (dx + 1)];
        }
    out_ws[idx] += acc;
}

// ---------------------------------------------------------------------------
extern "C" void kernel_launch(void* const* d_in, const int* in_sizes, int n_in,
                              void* d_out, int out_size, void* d_ws, size_t ws_size,
                              hipStream_t stream) {
    const float* x      = (const float*)d_in[0];
    const float* qkv_w  = (const float*)d_in[1];
    const float* qkv_b  = (const float*)d_in[2];
    const float* proj_w = (const float*)d_in[3];
    const float* proj_b = (const float*)d_in[4];
    const float* pe_w   = (const float*)d_in[5];
    const float* pe_b   = (const float*)d_in[6];
    const float* g1_w   = (const float*)d_in[7];
    const float* g1_b   = (const float*)d_in[8];
    const float* g2_w   = (const float*)d_in[9];
    const float* g2_b   = (const float*)d_in[10];

    // workspace carve-up (floats)
    int*   dynk    = (int*)d_ws;                           // [0]
    float* gate_ws = (float*)d_ws + 16;                    // 4096
    float* qkv_ws  = gate_ws + NB * NPIX;                  // 4*512*1024
    float* out_ws  = qkv_ws + (size_t)NB * QKV_OUT * NPIX; // 4*256*1024

    k_gate<<<NB * NPIX, 128, 0, stream>>>(x, g1_w, g1_b, g2_w, g2_b, gate_ws);
    k_dynk<<<1, 256, 0, stream>>>(gate_ws, dynk);
    k_gemm<<<dim3(QKV_OUT / 64, NPIX / 16, NB), 128, 0, stream>>>(x, qkv_w, qkv_b, qkv_ws);
    k_attn<<<dim3(NPIX / 16, NHEAD, NB), 128, 0, stream>>>(qkv_ws, dynk, out_ws);
    k_pe<<<(NB * DIM * NPIX + 255) / 256, 256, 0, stream>>>(qkv_ws, pe_w, pe_b, out_ws);
    k_gemm<<<dim3(DIM / 64, NPIX / 16, NB), 128, 0, stream>>>(out_ws, proj_w, proj_b,
                                                              (float*)d_out);
}